// ppro_cd_loss_88038239634160
// MI455X (gfx1250) — compile-verified
//
#include <hip/hip_runtime.h>
#include <hip/hip_bf16.h>
#include <math.h>

typedef __attribute__((ext_vector_type(2))) float v2f;
typedef __attribute__((ext_vector_type(8))) float v8f;

// ---------------------------------------------------------------------------
// FPS: one workgroup per batch. 8192 points, 256 threads, 32 points/thread
// held in registers (coords + running dist_min). 4096 sequential steps.
// Per step: per-thread min-update + argmax, wave32 shfl_xor max-reduce on a
// packed (dist_bits << 32 | 8191-idx) key, then ONE ds_max_u64 per wave into a
// rotating LDS slot + single barrier. Packed key gives jnp.argmax semantics
// (max value, first index on ties) since distances are non-negative floats.
// Slot rotation over 4 entries makes the reset race-free: iteration t writes
// slot[t&3] and resets slot[(t+2)&3], whose last readers were separated from
// this point by a full barrier.
// ---------------------------------------------------------------------------
#define FPS_N   8192
#define FPS_T   256
#define FPS_PPT 32
#define FPS_S   4096

__global__ void __launch_bounds__(FPS_T) fps_kernel(const float* __restrict__ gt,
                                                    float* __restrict__ out) {
  const int b    = blockIdx.x;
  const int tid  = threadIdx.x;
  const int lane = tid & 31;
  const float* pts  = gt  + (size_t)b * FPS_N * 3;
  float*       outp = out + (size_t)b * FPS_S * 3;

  float px[FPS_PPT], py[FPS_PPT], pz[FPS_PPT], dmin[FPS_PPT];
#pragma unroll
  for (int k = 0; k < FPS_PPT; ++k) {
    const int idx = tid + k * FPS_T;           // strided -> coalesced load
    px[k] = pts[idx * 3 + 0];
    py[k] = pts[idx * 3 + 1];
    pz[k] = pts[idx * 3 + 2];
    dmin[k] = 1.0e10f;                          // BIG
  }

  __shared__ unsigned long long s_slot[4];
  if (tid < 4) s_slot[tid] = 0ULL;
  __syncthreads();

  int far = 0;
  for (int t = 0; t < FPS_S; ++t) {
    // centroid = pts[far]; it IS sample t (uniform address -> broadcast load).
    const float cx = pts[far * 3 + 0];
    const float cy = pts[far * 3 + 1];
    const float cz = pts[far * 3 + 2];
    if (tid == 0) {
      outp[t * 3 + 0] = cx;
      outp[t * 3 + 1] = cy;
      outp[t * 3 + 2] = cz;
    }
    if (t == FPS_S - 1) break;   // last step's argmax is never consumed

    float bv = -1.0f;
    int   bi = 0;
#pragma unroll
    for (int k = 0; k < FPS_PPT; ++k) {
      const float dx = px[k] - cx, dy = py[k] - cy, dz = pz[k] - cz;
      const float d  = dx * dx + dy * dy + dz * dz;
      const float dm = fminf(dmin[k], d);
      dmin[k] = dm;
      if (dm > bv) { bv = dm; bi = tid + k * FPS_T; }  // ascending idx per thread
    }

    // Pack: non-negative float bits are order-preserving; complement index so
    // u64 max == (max dist, min idx).
    unsigned long long key =
        ((unsigned long long)__float_as_uint(bv) << 32) |
        (unsigned long long)(unsigned)(FPS_N - 1 - bi);

    // wave32 max-reduce on packed key
#pragma unroll
    for (int m = 16; m >= 1; m >>= 1) {
      const unsigned long long o = __shfl_xor(key, m, 32);
      if (o > key) key = o;
    }
    if (lane == 0) atomicMax(&s_slot[t & 3], key);      // ds_max_u64, 8 waves
    if (tid == 0) s_slot[(t + 2) & 3] = 0ULL;           // rotate-reset (safe)
    __syncthreads();
    const unsigned long long win = s_slot[t & 3];
    far = FPS_N - 1 - (int)(unsigned)(win & 0xffffffffULL);
  }
}

// ---------------------------------------------------------------------------
// Chamfer row-min via V_WMMA_F32_16X16X4_F32.
// One wave per 16-row A tile. A-frag (16x4, K=4, coord pad 0):
//   lanes 0-15 : VGPR0=K0=x, VGPR1=K1=y   (row = lane&15)
//   lanes16-31 : VGPR0=K2=z, VGPR1=K3=0
// B-frag mirrors it (col = lane&15). C/D f32 16x16: lane l, reg v -> row
// v+8*(l>=16), col l&15. d(i,j) = aa_i + bb_j - 2*ab -> fold into 8 row-mins,
// then shfl_xor reduce over the 16 columns in each half.
// ---------------------------------------------------------------------------
__global__ void __launch_bounds__(256) chamfer_rowmin_wmma(
    const float* __restrict__ A, int nA, int sA,
    const float* __restrict__ Bm, int nB, int sB,
    float* __restrict__ outMin) {
  const int lane   = threadIdx.x & 31;
  const int wave   = threadIdx.x >> 5;
  const int tile   = blockIdx.x * (blockDim.x >> 5) + wave;
  const int nTiles = nA >> 4;
  if (tile >= nTiles) return;                 // wave-uniform; EXEC stays all-1s
  const int b      = blockIdx.y;
  const int half   = lane >> 4;
  const int lane15 = lane & 15;

  const float* ap = A + (size_t)b * sA + (size_t)(tile * 16 + lane15) * 3;
  const float ax = ap[0], ay = ap[1], az = ap[2];
  const float aa = ax * ax + ay * ay + az * az;

  v2f afrag;
  afrag.x = half ? az : ax;
  afrag.y = half ? 0.0f : ay;

  float aa_r[8];
#pragma unroll
  for (int v = 0; v < 8; ++v)
    aa_r[v] = __shfl(aa, v + half * 8, 32);   // rows 0-7 (half 0) / 8-15 (half 1)

  float rmin[8];
#pragma unroll
  for (int v = 0; v < 8; ++v) rmin[v] = 3.0e38f;

  const float* bbase = Bm + (size_t)b * sB;
  const int nCT = nB >> 4;
  for (int ct = 0; ct < nCT; ++ct) {
    const float* bp = bbase + (size_t)(ct * 16 + lane15) * 3;
    const float bx = bp[0], by = bp[1], bz = bp[2];
    const float bb = bx * bx + by * by + bz * bz;

    v2f bfrag;
    bfrag.x = half ? bz : bx;
    bfrag.y = half ? 0.0f : by;

    v8f acc = {};
    acc = __builtin_amdgcn_wmma_f32_16x16x4_f32(
        /*neg_a=*/false, afrag, /*neg_b=*/false, bfrag,
        /*c_mod=*/(short)0, acc, /*reuse_a=*/false, /*reuse_b=*/false);

#pragma unroll
    for (int v = 0; v < 8; ++v) {
      const float d = __builtin_fmaf(-2.0f, acc[v], aa_r[v] + bb);
      rmin[v] = fminf(rmin[v], d);
    }
  }

  // min over the 16 columns held in each 16-lane half
#pragma unroll
  for (int m = 8; m >= 1; m >>= 1) {
#pragma unroll
    for (int v = 0; v < 8; ++v) {
      const float o = __shfl_xor(rmin[v], m, 32);
      rmin[v] = fminf(rmin[v], o);
    }
  }

  if (lane15 == 0) {
    float* op = outMin + (size_t)b * nA + tile * 16 + half * 8;
#pragma unroll
    for (int v = 0; v < 8; ++v)
      op[v] = sqrtf(fmaxf(rmin[v], 1e-9f));   // sqrt(clip(min, 1e-9))
  }
}

// ---------------------------------------------------------------------------
// Final deterministic reduction: fixed-stride per-thread sums + fixed LDS tree.
// ---------------------------------------------------------------------------
__global__ void __launch_bounds__(256) finalize_kernel(
    const float* __restrict__ m_p1, const float* __restrict__ m_g1,
    const float* __restrict__ m_p2, const float* __restrict__ m_g2,
    float* __restrict__ out) {
  __shared__ float red[256];
  const int tid = threadIdx.x;
  float s[4] = {0.f, 0.f, 0.f, 0.f};
  for (int i = tid; i < 8192; i += 256)  { s[0] += m_p1[i]; s[1] += m_g1[i]; }
  for (int i = tid; i < 32768; i += 256) { s[2] += m_p2[i]; s[3] += m_g2[i]; }

  float tot[4];
  for (int a = 0; a < 4; ++a) {
    red[tid] = s[a];
    __syncthreads();
    for (int off = 128; off >= 1; off >>= 1) {
      if (tid < off) red[tid] += red[tid + off];
      __syncthreads();
    }
    tot[a] = red[0];
    __syncthreads();
  }
  if (tid == 0) {
    const float cd1 = 0.5f * (tot[0] * (1.0f / 8192.0f)  + tot[1] * (1.0f / 8192.0f));
    const float cd2 = 0.5f * (tot[2] * (1.0f / 32768.0f) + tot[3] * (1.0f / 32768.0f));
    out[0] = (cd1 + cd2) * 1000.0f;
    out[1] = cd1 * 1000.0f;
    out[2] = cd2 * 1000.0f;
  }
}

// ---------------------------------------------------------------------------
// Launch: FPS(gt,4096) once per batch (gt1 = its first 1024 rows), 4 WMMA
// chamfer passes, then finalize. All scratch in d_ws, fully overwritten.
// ---------------------------------------------------------------------------
extern "C" void kernel_launch(void* const* d_in, const int* in_sizes, int n_in,
                              void* d_out, int out_size, void* d_ws, size_t ws_size,
                              hipStream_t stream) {
  (void)in_sizes; (void)n_in; (void)out_size; (void)ws_size;
  const float* p1 = (const float*)d_in[0];   // [8,1024,3]
  const float* p2 = (const float*)d_in[1];   // [8,4096,3]
  const float* gt = (const float*)d_in[2];   // [8,8192,3]
  float* out = (float*)d_out;                // 3 scalars

  float* ws   = (float*)d_ws;
  float* gt2  = ws;                          // 8*4096*3 = 98304 floats
  float* m_p1 = gt2 + 8 * 4096 * 3;          // 8192
  float* m_g1 = m_p1 + 8 * 1024;             // 8192
  float* m_p2 = m_g1 + 8 * 1024;             // 32768
  float* m_g2 = m_p2 + 8 * 4096;             // 32768

  // FPS: 8 batches in parallel, gt2 = sampled points; gt1 = first 1024 rows.
  fps_kernel<<<dim3(8), dim3(FPS_T), 0, stream>>>(gt, gt2);

  // cd1: p1 vs gt1 (gt1 rows live inside gt2 with batch stride 4096*3)
  chamfer_rowmin_wmma<<<dim3(8, 8), 256, 0, stream>>>(p1, 1024, 1024 * 3,
                                                      gt2, 1024, 4096 * 3, m_p1);
  chamfer_rowmin_wmma<<<dim3(8, 8), 256, 0, stream>>>(gt2, 1024, 4096 * 3,
                                                      p1, 1024, 1024 * 3, m_g1);
  // cd2: p2 vs gt2
  chamfer_rowmin_wmma<<<dim3(32, 8), 256, 0, stream>>>(p2, 4096, 4096 * 3,
                                                       gt2, 4096, 4096 * 3, m_p2);
  chamfer_rowmin_wmma<<<dim3(32, 8), 256, 0, stream>>>(gt2, 4096, 4096 * 3,
                                                       p2, 4096, 4096 * 3, m_g2);

  finalize_kernel<<<1, 256, 0, stream>>>(m_p1, m_g1, m_p2, m_g2, out);
}